// AggregationGNN_19980187861090
// MI455X (gfx1250) — compile-verified
//
#include <hip/hip_runtime.h>

typedef float v2f __attribute__((ext_vector_type(2)));
typedef float v8f __attribute__((ext_vector_type(8)));

// ---------------------------------------------------------------------------
// Zero the [N,64] f32 aggregation scratch (float4 stores, grid-stride).
// ---------------------------------------------------------------------------
__global__ void gnn_zero(float4* p, long n4) {
    long i = (long)blockIdx.x * blockDim.x + threadIdx.x;
    long stride = (long)gridDim.x * blockDim.x;
    float4 z = make_float4(0.f, 0.f, 0.f, 0.f);
    for (; i < n4; i += stride) p[i] = z;
}

// ---------------------------------------------------------------------------
// Edge phase: bond embed (RBF + categorical) + gather src feats + atomic
// scatter-add into agg[dst]. One wave (32 lanes) per edge, lane = feature.
// ---------------------------------------------------------------------------
__global__ void gnn_edge(const float* __restrict__ atom,
                         const int* __restrict__ src, const int* __restrict__ dst,
                         const int* __restrict__ rdir, const int* __restrict__ rtyp,
                         const int* __restrict__ rring,
                         const int* __restrict__ pdir, const int* __restrict__ ptyp,
                         const int* __restrict__ pring,
                         const float* __restrict__ rlen, const float* __restrict__ plen,
                         const float* __restrict__ emb_dir, const float* __restrict__ emb_type,
                         const float* __restrict__ emb_ring,
                         const float* __restrict__ rbfW, const float* __restrict__ rbfb,
                         float* __restrict__ agg, int E) {
    __shared__ float s_dir[8 * 32];
    __shared__ float s_typ[16 * 32];
    __shared__ float s_rng[4 * 32];
    __shared__ float s_W[20 * 32];
    __shared__ float s_b[32];
    for (int i = threadIdx.x; i < 8 * 32;  i += blockDim.x) s_dir[i] = emb_dir[i];
    for (int i = threadIdx.x; i < 16 * 32; i += blockDim.x) s_typ[i] = emb_type[i];
    for (int i = threadIdx.x; i < 4 * 32;  i += blockDim.x) s_rng[i] = emb_ring[i];
    for (int i = threadIdx.x; i < 20 * 32; i += blockDim.x) s_W[i]   = rbfW[i];
    for (int i = threadIdx.x; i < 32;      i += blockDim.x) s_b[i]   = rbfb[i];
    __syncthreads();

    const int lane = threadIdx.x & 31;
    const int wid  = threadIdx.x >> 5;
    const int wpb  = blockDim.x >> 5;
    const int gw   = blockIdx.x * wpb + wid;
    const int nw   = gridDim.x * wpb;

    for (int e = gw; e < E; e += nw) {
        const int s  = src[e];
        const int dn = dst[e];
        const float rl = rlen[e];
        const float pl = plen[e];

        float catr = s_dir[rdir[e] * 32 + lane] + s_typ[rtyp[e] * 32 + lane] +
                     s_rng[rring[e] * 32 + lane];
        float catp = s_dir[pdir[e] * 32 + lane] + s_typ[ptyp[e] * 32 + lane] +
                     s_rng[pring[e] * 32 + lane];

        // lanes 0..19 each evaluate one RBF center; broadcast via shfl
        float rvr = 0.f, rvp = 0.f;
        if (lane < 20) {
            float c  = 0.1f * (float)lane;
            float dr = rl - c;
            float dp = pl - c;
            rvr = __expf(-10.0f * dr * dr);
            rvp = __expf(-10.0f * dp * dp);
        }
        float accr = catr + s_b[lane];
        float accp = catp + s_b[lane];
#pragma unroll
        for (int k = 0; k < 20; ++k) {
            float w = s_W[k * 32 + lane];
            accr = fmaf(__shfl(rvr, k, 32), w, accr);
            accp = fmaf(__shfl(rvp, k, 32), w, accp);
        }

        const long sb = (long)s * 64;
        float lo = atom[sb + lane]      + accr;            // r_embed half
        float hi = atom[sb + 32 + lane] + (accp - accr);   // (p - r) half

        float* ab = agg + (long)dn * 64;
        atomicAdd(ab + lane,      lo);
        atomicAdd(ab + 32 + lane, hi);
    }
}

// ---------------------------------------------------------------------------
// Node phase: out = relu(relu(agg @ W1 + b1) @ W2 + b2), WMMA f32 16x16x4.
// 4 waves / 128 threads per block; each wave owns one 16-node tile.
// Weights live in LDS, (k,k+1)-interleaved so each B fragment is one b64 load.
// h tile bounces through per-wave LDS (same-wave LDS is in-order).
// ---------------------------------------------------------------------------
__global__ void __launch_bounds__(128)
gnn_node(const float* __restrict__ agg,
         const float* __restrict__ W1, const float* __restrict__ b1,
         const float* __restrict__ W2, const float* __restrict__ b2,
         float* __restrict__ out, int ntiles) {
    __shared__ float sW1[64 * 128];     // paired: [(k>>1)*128 + n]*2 + (k&1)
    __shared__ float sW2[128 * 64];     // paired: [(k>>1)*64  + n]*2 + (k&1)
    __shared__ float sb1[128];
    __shared__ float sb2[64];
    __shared__ float hbuf[4][16 * 132]; // per-wave 16x128 h tile, padded stride

    for (int i = threadIdx.x; i < 64 * 128; i += blockDim.x) {
        int k = i >> 7, n = i & 127;
        sW1[((k >> 1) * 128 + n) * 2 + (k & 1)] = W1[i];
    }
    for (int i = threadIdx.x; i < 128 * 64; i += blockDim.x) {
        int k = i >> 6, n = i & 63;
        sW2[((k >> 1) * 64 + n) * 2 + (k & 1)] = W2[i];
    }
    for (int i = threadIdx.x; i < 128; i += blockDim.x) sb1[i] = b1[i];
    for (int i = threadIdx.x; i < 64;  i += blockDim.x) sb2[i] = b2[i];
    __syncthreads();

    const int lane = threadIdx.x & 31;
    const int wid  = threadIdx.x >> 5;
    const int tile = blockIdx.x * 4 + wid;
    if (tile >= ntiles) return;          // wave-uniform: EXEC stays all-1s

    const int half = lane >> 4;          // 0: K+{0,1}, 1: K+{2,3}
    const int lr   = lane & 15;          // row (A/D) or col (B/D) within tile
    const long rowbase = (long)tile * 16;

    // ---- A fragments for layer 1: agg rows, K = 0..63 in chunks of 4 ----
    const float* arow = agg + (rowbase + lr) * 64 + 2 * half;
    v2f a[16];
#pragma unroll
    for (int kc = 0; kc < 16; ++kc)
        a[kc] = *(const v2f*)(arow + 4 * kc);

    float* hb = &hbuf[wid][0];

    // ---- layer 1: h = relu(agg @ W1 + b1), 8 col-tiles of 16 ----
#pragma unroll
    for (int nt = 0; nt < 8; ++nt) {
        v8f acc = {0.f, 0.f, 0.f, 0.f, 0.f, 0.f, 0.f, 0.f};
#pragma unroll
        for (int kc = 0; kc < 16; ++kc) {
            v2f b = *(const v2f*)&sW1[((2 * kc + half) * 128 + nt * 16 + lr) * 2];
            acc = __builtin_amdgcn_wmma_f32_16x16x4_f32(
                false, a[kc], false, b, (short)0, acc, false, false);
        }
        float bb = sb1[nt * 16 + lr];
#pragma unroll
        for (int v = 0; v < 8; ++v) {
            float x = acc[v] + bb;
            x = x > 0.f ? x : 0.f;
            hb[(v + 8 * half) * 132 + nt * 16 + lr] = x;   // D: row=v(+8), col=lr
        }
    }

    // ---- layer 2: out = relu(h @ W2 + b2), 4 col-tiles of 16 ----
#pragma unroll
    for (int nt = 0; nt < 4; ++nt) {
        v8f acc = {0.f, 0.f, 0.f, 0.f, 0.f, 0.f, 0.f, 0.f};
#pragma unroll
        for (int kc = 0; kc < 32; ++kc) {
            v2f a2 = *(const v2f*)&hb[lr * 132 + 4 * kc + 2 * half];
            v2f b  = *(const v2f*)&sW2[((2 * kc + half) * 64 + nt * 16 + lr) * 2];
            acc = __builtin_amdgcn_wmma_f32_16x16x4_f32(
                false, a2, false, b, (short)0, acc, false, false);
        }
        float bb = sb2[nt * 16 + lr];
#pragma unroll
        for (int v = 0; v < 8; ++v) {
            float x = acc[v] + bb;
            x = x > 0.f ? x : 0.f;
            out[(rowbase + v + 8 * half) * 64 + nt * 16 + lr] = x;
        }
    }
}

// ---------------------------------------------------------------------------
extern "C" void kernel_launch(void* const* d_in, const int* in_sizes, int n_in,
                              void* d_out, int out_size, void* d_ws, size_t ws_size,
                              hipStream_t stream) {
    const float* atom    = (const float*)d_in[0];
    const int*   src     = (const int*)d_in[1];
    const int*   dst     = (const int*)d_in[2];
    const int*   rdir    = (const int*)d_in[3];
    const int*   rtyp    = (const int*)d_in[4];
    const int*   rring   = (const int*)d_in[5];
    const int*   pdir    = (const int*)d_in[6];
    const int*   ptyp    = (const int*)d_in[7];
    const int*   pring   = (const int*)d_in[8];
    const float* rlen    = (const float*)d_in[9];
    const float* plen    = (const float*)d_in[10];
    const float* emb_dir = (const float*)d_in[11];
    const float* emb_typ = (const float*)d_in[12];
    const float* emb_rng = (const float*)d_in[13];
    const float* rbfW    = (const float*)d_in[14];
    const float* rbfb    = (const float*)d_in[15];
    const float* W1      = (const float*)d_in[16];
    const float* b1      = (const float*)d_in[17];
    const float* W2      = (const float*)d_in[18];
    const float* b2      = (const float*)d_in[19];

    const int N = in_sizes[0] / 64;   // CD = 64
    const int E = in_sizes[1];

    float* agg = (float*)d_ws;        // [N, 64] f32 accumulator
    const long n4 = (long)N * 64 / 4;

    gnn_zero<<<2048, 256, 0, stream>>>((float4*)agg, n4);

    gnn_edge<<<4096, 256, 0, stream>>>(atom, src, dst, rdir, rtyp, rring,
                                       pdir, ptyp, pring, rlen, plen,
                                       emb_dir, emb_typ, emb_rng, rbfW, rbfb,
                                       agg, E);

    const int ntiles = (N + 15) / 16;
    gnn_node<<<(ntiles + 3) / 4, 128, 0, stream>>>(agg, W1, b1, W2, b2,
                                                   (float*)d_out, ntiles);
}